// Regressor_7000796693089
// MI455X (gfx1250) — compile-verified
//
#include <hip/hip_runtime.h>
#include <math.h>

#define HW   4096
#define CC   235
#define NJ   21
#define NB   128
#define KB   64
#define LDSS 68            // padded LDS row stride (floats): bank-conflict-free b64 frag reads
#define HF   128           // heads*fo
#define NE   41

typedef __attribute__((ext_vector_type(2))) float v2f;
typedef __attribute__((ext_vector_type(8))) float v8f;

// ---- CDNA5 async copy helpers ---------------------------------------------
__device__ __forceinline__ void async_ld128(unsigned ldsoff, const float* gp) {
    // GLOBAL_LOAD_ASYNC_TO_LDS_B128 (GV mode): vdst = per-lane LDS byte addr,
    // vaddr = per-lane 64-bit global addr. Tracked by ASYNCcnt.
    asm volatile("global_load_async_to_lds_b128 %0, %1, off"
                 :: "v"(ldsoff), "v"((unsigned long long)(uintptr_t)gp)
                 : "memory");
}

#if __has_builtin(__builtin_amdgcn_s_wait_asynccnt)
#define WAIT_ASYNC(n) __builtin_amdgcn_s_wait_asynccnt(n)
#else
#define WAIT_ASYNC(n) asm volatile("s_wait_asynccnt %0" :: "i"(n) : "memory")
#endif

// ---------------------------------------------------------------------------
// Kernel 1: q[b,n,c] = sum_p t[b,n,p] * f[b,c,p]   (memory-bound, WMMA f32)
// grid = (15 channel tiles, 128 batches), block = 64 threads (2 waves)
// Double-buffered LDS filled by global_load_async_to_lds_b128 (ASYNCcnt),
// WMMAs on buffer A overlap async streaming into buffer B.
// ---------------------------------------------------------------------------
__global__ __launch_bounds__(64) void qgemm_kernel(const float* __restrict__ f,
                                                   const float* __restrict__ t,
                                                   float* __restrict__ q)
{
    __shared__ float fS[2][16 * LDSS];
    __shared__ float tS[2][32 * LDSS];

    const int b   = blockIdx.y;
    const int c0  = blockIdx.x * 16;
    const int tid = threadIdx.x;
    const int lane = tid & 31;
    const int wave = tid >> 5;      // M-tile index
    const int kh   = lane >> 4;     // K half (0: K=0,1  1: K=2,3)
    const int ln   = lane & 15;

    const float* __restrict__ fB = f + (size_t)b * CC * HW;
    const float* __restrict__ tB = t + (size_t)b * NJ * HW;

    // per-thread staging coordinates (uniform trip counts -> 10 issues/wave/chunk)
    // f tile: 256 float4 over 64 threads = 4 issues; t tile padded to 384 = 6 issues
    v8f acc0 = {};
    v8f acc1 = {};

    const int arow = (wave * 16 + ln) * LDSS + 2 * kh;
    const int brow = ln * LDSS + 2 * kh;

    // stage chunk 0 into buffer 0
    {
        const int kb = 0;
        for (int e = tid; e < 256; e += 64) {
            int row = e >> 4, col = e & 15;
            int cg = c0 + row; if (cg > CC - 1) cg = CC - 1;
            async_ld128((unsigned)(uintptr_t)&fS[0][row * LDSS + col * 4],
                        fB + (size_t)cg * HW + kb + col * 4);
        }
        for (int e = tid; e < 384; e += 64) {
            int row = e >> 4, col = e & 15;
            int rg = row > NJ - 1 ? NJ - 1 : row;       // clamp global source only
            async_ld128((unsigned)(uintptr_t)&tS[0][row * LDSS + col * 4],
                        tB + (size_t)rg * HW + kb + col * 4);
        }
    }

    for (int kb = 0; kb < HW; kb += KB) {
        const int cur = (kb >> 6) & 1;
        const int nxt = cur ^ 1;
        if (kb + KB < HW) {
            // stream next chunk while this one is consumed
            for (int e = tid; e < 256; e += 64) {
                int row = e >> 4, col = e & 15;
                int cg = c0 + row; if (cg > CC - 1) cg = CC - 1;
                async_ld128((unsigned)(uintptr_t)&fS[nxt][row * LDSS + col * 4],
                            fB + (size_t)cg * HW + kb + KB + col * 4);
            }
            for (int e = tid; e < 384; e += 64) {
                int row = e >> 4, col = e & 15;
                int rg = row > NJ - 1 ? NJ - 1 : row;
                async_ld128((unsigned)(uintptr_t)&tS[nxt][row * LDSS + col * 4],
                            tB + (size_t)rg * HW + kb + KB + col * 4);
            }
            WAIT_ASYNC(10);   // drain current chunk's 10; next chunk's 10 in flight
        } else {
            WAIT_ASYNC(0);    // last chunk: drain everything
        }
        __syncthreads();      // all waves' async writes for 'cur' visible

        const float* aP = &tS[cur][arow];
        const float* bP = &fS[cur][brow];
        #pragma unroll
        for (int k = 0; k < KB; k += 8) {
            v2f a0, b0, a1, b1;
            a0.x = aP[k];     a0.y = aP[k + 1];
            b0.x = bP[k];     b0.y = bP[k + 1];
            acc0 = __builtin_amdgcn_wmma_f32_16x16x4_f32(false, a0, false, b0,
                                                         (short)0, acc0, false, false);
            a1.x = aP[k + 4]; a1.y = aP[k + 5];
            b1.x = bP[k + 4]; b1.y = bP[k + 5];
            acc1 = __builtin_amdgcn_wmma_f32_16x16x4_f32(false, a1, false, b1,
                                                         (short)0, acc1, false, false);
        }
        __syncthreads();      // 'cur' fully consumed before it is refilled (iter+1)
    }

    v8f acc = acc0 + acc1;
    #pragma unroll
    for (int v = 0; v < 8; v++) {
        int m = wave * 16 + v + 8 * kh;   // C/D layout: lanes 16-31 hold M = v+8
        int c = c0 + ln;
        if (m < NJ && c < CC)
            q[((size_t)b * NJ + m) * CC + c] = acc[v];
    }
}

// ---------------------------------------------------------------------------
// Kernel 2: BatchNorm statistics per joint over (B, C) = 30080 elements
// ---------------------------------------------------------------------------
__global__ __launch_bounds__(256) void bnstats_kernel(const float* __restrict__ q,
                                                      float* __restrict__ meanA,
                                                      float* __restrict__ istdA)
{
    const int n = blockIdx.x, tid = threadIdx.x;
    float s = 0.f, s2 = 0.f;
    for (int i = tid; i < NB * CC; i += 256) {
        int b = i / CC, c = i - b * CC;
        float v = q[((size_t)b * NJ + n) * CC + c];
        s += v; s2 += v * v;
    }
    __shared__ float rs[256], rs2[256];
    rs[tid] = s; rs2[tid] = s2;
    __syncthreads();
    for (int off = 128; off > 0; off >>= 1) {
        if (tid < off) { rs[tid] += rs[tid + off]; rs2[tid] += rs2[tid + off]; }
        __syncthreads();
    }
    if (tid == 0) {
        const float M = (float)(NB * CC);
        float mean = rs[0] / M;
        float var  = rs2[0] / M - mean * mean;
        meanA[n] = mean;
        istdA[n] = rsqrtf(var + 1e-5f);
    }
}

// ---------------------------------------------------------------------------
// Kernel 3a: in-place BN + LeakyReLU(0.1)
// ---------------------------------------------------------------------------
__global__ __launch_bounds__(256) void bnorm_kernel(float* __restrict__ q,
                                                    const float* __restrict__ meanA,
                                                    const float* __restrict__ istdA,
                                                    const float* __restrict__ gamma,
                                                    const float* __restrict__ beta)
{
    size_t i = (size_t)blockIdx.x * 256 + threadIdx.x;
    if (i >= (size_t)NB * NJ * CC) return;
    int n = (int)((i / CC) % NJ);
    float v = (q[i] - meanA[n]) * istdA[n] * gamma[n] + beta[n];
    q[i] = v >= 0.f ? v : 0.1f * v;
}

// ---------------------------------------------------------------------------
// Kernel 3b: h = qn @ gat_w   [2688,235]x[235,128]  (tiny, L2 resident)
// ---------------------------------------------------------------------------
__global__ __launch_bounds__(256) void hgemm_kernel(const float* __restrict__ qn,
                                                    const float* __restrict__ W,
                                                    float* __restrict__ h)
{
    int idx = blockIdx.x * 256 + threadIdx.x;
    if (idx >= NB * NJ * HF) return;
    int row = idx >> 7, col = idx & 127;
    const float* qr = qn + (size_t)row * CC;
    float s = 0.f;
    for (int k = 0; k < CC; k++) s += qr[k] * W[k * HF + col];
    h[idx] = s;
}

// ---------------------------------------------------------------------------
// Kernel 4: per-batch GAT (softmax attention over 41 edges) + ChebConv K=2
// one block (128 threads) per batch sample
// ---------------------------------------------------------------------------
__global__ __launch_bounds__(128) void graph_kernel(const float* __restrict__ h,
                                                    const float* __restrict__ att_src,
                                                    const float* __restrict__ att_dst,
                                                    const float* __restrict__ gat_bias,
                                                    const float* __restrict__ cheb_w,
                                                    const float* __restrict__ cheb_bias,
                                                    const float* __restrict__ adj,
                                                    const int* __restrict__ esrc,
                                                    const int* __restrict__ edst,
                                                    float* __restrict__ y)
{
    const int b = blockIdx.x, tid = threadIdx.x;
    __shared__ float hS[NJ][HF];
    __shared__ float xS[NJ][HF];
    __shared__ float asrc[NJ][4], adst[NJ][4];
    __shared__ float ew[NE][4];
    __shared__ float mx[NJ][4], den[NJ][4];
    __shared__ float dS[NJ];
    __shared__ float z0[NJ][3], z1[NJ][3];
    __shared__ int   es[NE], ed[NE];

    const float* hb = h + (size_t)b * NJ * HF;
    for (int i = tid; i < NJ * HF; i += 128) hS[i / HF][i % HF] = hb[i];
    if (tid < NE) { es[tid] = esrc[tid]; ed[tid] = edst[tid]; }
    if (tid < NJ) {
        float s = 0.f;
        for (int j = 0; j < NJ; j++) s += adj[tid * NJ + j];
        dS[tid] = rsqrtf(s);
    }
    __syncthreads();

    if (tid < NJ * 4) {                       // attention logits per node/head
        int n = tid >> 2, hd = tid & 3;
        float s = 0.f, d2 = 0.f;
        for (int fo = 0; fo < 32; fo++) {
            float v = hS[n][hd * 32 + fo];
            s  += v * att_src[hd * 32 + fo];
            d2 += v * att_dst[hd * 32 + fo];
        }
        asrc[n][hd] = s; adst[n][hd] = d2;
    }
    __syncthreads();

    for (int i = tid; i < NE * 4; i += 128) { // edge scores, leaky 0.2
        int e = i >> 2, hd = i & 3;
        float v = asrc[es[e]][hd] + adst[ed[e]][hd];
        ew[e][hd] = v >= 0.f ? v : 0.2f * v;
    }
    __syncthreads();

    if (tid < NJ * 4) {                       // per-dst max
        int n = tid >> 2, hd = tid & 3;
        float m = -1e30f;
        for (int e = 0; e < NE; e++) if (ed[e] == n) m = fmaxf(m, ew[e][hd]);
        mx[n][hd] = m;
    }
    __syncthreads();
    if (tid < NJ * 4) {                       // per-dst denom
        int n = tid >> 2, hd = tid & 3;
        float s = 0.f;
        for (int e = 0; e < NE; e++) if (ed[e] == n) s += __expf(ew[e][hd] - mx[n][hd]);
        den[n][hd] = s;
    }
    __syncthreads();
    for (int i = tid; i < NE * 4; i += 128) { // alpha
        int e = i >> 2, hd = i & 3, n = ed[e];
        ew[e][hd] = __expf(ew[e][hd] - mx[n][hd]) / den[n][hd];
    }
    __syncthreads();

    {                                          // aggregate + bias
        int c = tid, hd = c >> 5;
        for (int n = 0; n < NJ; n++) {
            float s = 0.f;
            for (int e = 0; e < NE; e++)
                if (ed[e] == n) s += ew[e][hd] * hS[es[e]][c];
            xS[n][c] = s + gat_bias[c];
        }
    }
    __syncthreads();

    if (tid < 2 * NJ * 3) {                   // z_k = x @ cheb_w[k]
        int kk = tid / (NJ * 3);
        int r  = tid % (NJ * 3);
        int m = r / 3, o = r % 3;
        const float* w = cheb_w + kk * HF * 3;
        float s = 0.f;
        for (int c = 0; c < HF; c++) s += xS[m][c] * w[c * 3 + o];
        if (kk) z1[m][o] = s; else z0[m][o] = s;
    }
    __syncthreads();

    if (tid < NJ * 3) {                       // y = z0 + L @ z1 + bias
        int m = tid / 3, o = tid % 3;
        float s = z0[m][o];
        for (int n = 0; n < NJ; n++) {
            float L = (m == n ? 1.f : 0.f) - dS[m] * adj[m * NJ + n] * dS[n];
            s += L * z1[n][o];
        }
        y[((size_t)b * NJ + m) * 3 + o] = s + cheb_bias[o];
    }
}

// ---------------------------------------------------------------------------
extern "C" void kernel_launch(void* const* d_in, const int* in_sizes, int n_in,
                              void* d_out, int out_size, void* d_ws, size_t ws_size,
                              hipStream_t stream)
{
    const float* f        = (const float*)d_in[0];
    const float* t        = (const float*)d_in[1];
    const float* gamma    = (const float*)d_in[2];
    const float* beta     = (const float*)d_in[3];
    const float* gat_w    = (const float*)d_in[4];
    const float* att_src  = (const float*)d_in[5];
    const float* att_dst  = (const float*)d_in[6];
    const float* gat_bias = (const float*)d_in[7];
    const float* cheb_w   = (const float*)d_in[8];
    const float* cheb_b   = (const float*)d_in[9];
    const float* adj      = (const float*)d_in[10];
    const int*   esrc     = (const int*)d_in[11];
    const int*   edst     = (const int*)d_in[12];
    float* out = (float*)d_out;

    char* ws = (char*)d_ws;
    float* q     = (float*)ws;                      // 128*21*235 f32 = 2,526,720 B
    float* meanA = (float*)(ws + 2526720);          // 21 f32
    float* istdA = meanA + 32;                      // 21 f32
    float* h     = (float*)(ws + 2527232);          // 2688*128 f32 = 1,376,256 B

    qgemm_kernel  <<<dim3(15, NB), 64,  0, stream>>>(f, t, q);
    bnstats_kernel<<<NJ,           256, 0, stream>>>(q, meanA, istdA);
    bnorm_kernel  <<<(NB * NJ * CC + 255) / 256, 256, 0, stream>>>(q, meanA, istdA, gamma, beta);
    hgemm_kernel  <<<(NB * NJ * HF + 255) / 256, 256, 0, stream>>>(q, gat_w, h);
    graph_kernel  <<<NB, 128, 0, stream>>>(h, att_src, att_dst, gat_bias,
                                           cheb_w, cheb_b, adj, esrc, edst, out);
}